// GraphEncoder_17721035063879
// MI455X (gfx1250) — compile-verified
//
#include <hip/hip_runtime.h>
#include <math.h>

typedef __attribute__((ext_vector_type(2))) float v2f;
typedef __attribute__((ext_vector_type(8))) float v8f;

#define NFEAT 768
#define NEG_SLOPE 0.2f
#define KC 64          // K-chunk staged in LDS
#define ROWS 80        // rows per block: 20000 = 250 * 80
#define LDS_STRIDE 68  // padded row stride (floats): conflict-free + 8B aligned

// ---------------------------------------------------------------------------
// Async copy: global -> LDS, 8 bytes per lane, tracked by ASYNCcnt.
// VDST holds the workgroup-relative LDS byte offset (low 32 bits of the
// generic pointer to a __shared__ object); VADDR holds the 64-bit global
// address. (CDNA5 ISA 10.7 / 15.18.3, GLOBAL_LOAD_ASYNC_TO_LDS_B64.)
// ---------------------------------------------------------------------------
__device__ __forceinline__ void async_copy_b64(const float* gp, float* lp)
{
    const unsigned ldsoff = (unsigned)(size_t)lp;
    const unsigned long long ga = (unsigned long long)(size_t)gp;
    asm volatile("global_load_async_to_lds_b64 %0, %1, off"
                 :: "v"(ldsoff), "v"(ga) : "memory");
}

// Stage one 80x64 A-chunk: 2560 b64 transfers, exactly 5 per thread,
// so every wave issues exactly 5 async ops per stage.
__device__ __forceinline__ void stage_chunk(float* buf, const float* A,
                                            int m0, int kc, int tid)
{
#pragma unroll
    for (int j = 0; j < 5; ++j) {
        const int i   = tid + j * 512;   // < 2560, uniform trip count
        const int row = i >> 5;          // 32 transfers per 64-float row
        const int c2  = (i & 31) << 1;
        async_copy_b64(A + (size_t)(m0 + row) * NFEAT + kc + c2,
                       buf + row * LDS_STRIDE + c2);
    }
}

// ---------------------------------------------------------------------------
// C[M,768] = A[M,768] @ B[768,768], fp32, V_WMMA_F32_16X16X4_F32.
// Block: 512 threads = 16 waves, covers 80 rows x 768 cols.
// Wave w: 80x48 strip at columns [48w, 48w+48): 5 M-subtiles x 3 N-subtiles.
// A double-buffered through LDS via async-to-LDS DMA; B read from global
// (L2-resident; each block reads B once).
// ---------------------------------------------------------------------------
__global__ __launch_bounds__(512) void gemm_wmma_f32(
    const float* __restrict__ A, const float* __restrict__ B,
    float* __restrict__ C)
{
    __shared__ float As[2][ROWS * LDS_STRIDE];  // ~42.5 KB total

    const int m0   = blockIdx.x * ROWS;
    const int tid  = threadIdx.x;
    const int w    = tid >> 5;
    const int lane = tid & 31;
    const int half = lane >> 4;  // 0: K=0,1  1: K=2,3 within a 4-wide K step
    const int ln   = lane & 15;
    const int cb   = w * 48;     // wave's column base

    v8f acc[5][3] = {};

    stage_chunk(As[0], A, m0, 0, tid);   // prefetch chunk 0

    int cur = 0;
    for (int kc = 0; kc < NFEAT; kc += KC) {
        const bool more = (kc + KC) < NFEAT;
        if (more) {
            stage_chunk(As[cur ^ 1], A, m0, kc + KC, tid);
            // 5 newly issued ops outstanding; current chunk's 5 retired.
            asm volatile("s_wait_asynccnt 0x5" ::: "memory");
        } else {
            asm volatile("s_wait_asynccnt 0x0" ::: "memory");
        }
        __syncthreads();  // all waves' DMA for this chunk complete

        const float* Abuf = As[cur];
#pragma unroll 2
        for (int ks = 0; ks < KC; ks += 4) {
            const int ka = ks + 2 * half;
            const float* Brow0 = B + (size_t)(kc + ka) * NFEAT + cb + ln;
            const float* Brow1 = Brow0 + NFEAT;
            v2f bf[3];
#pragma unroll
            for (int t = 0; t < 3; ++t) {
                bf[t].x = Brow0[t * 16];
                bf[t].y = Brow1[t * 16];
            }
#pragma unroll
            for (int mb = 0; mb < 5; ++mb) {
                v2f a;
                const float* ap = &Abuf[(mb * 16 + ln) * LDS_STRIDE + ka];
                a.x = ap[0];
                a.y = ap[1];
#pragma unroll
                for (int t = 0; t < 3; ++t) {
                    acc[mb][t] = __builtin_amdgcn_wmma_f32_16x16x4_f32(
                        false, a, false, bf[t], (short)0, acc[mb][t],
                        false, false);
                }
            }
        }
        __syncthreads();  // done reading As[cur]; safe to overwrite next round
        cur ^= 1;
    }

#pragma unroll
    for (int mb = 0; mb < 5; ++mb) {
#pragma unroll
        for (int t = 0; t < 3; ++t) {
            const int nb = cb + t * 16;
#pragma unroll
            for (int v = 0; v < 8; ++v) {
                C[(size_t)(m0 + mb * 16 + v + 8 * half) * NFEAT + nb + ln] =
                    acc[mb][t][v];
            }
        }
    }
}

// ---------------------------------------------------------------------------
// Per-node attention logits: ls[n,h] = <h[n,h,:], a_src[h,:]>, same for ld.
// ---------------------------------------------------------------------------
__global__ __launch_bounds__(256) void logits_kernel(
    const float* __restrict__ h, const float* __restrict__ a_s,
    const float* __restrict__ a_d, float* __restrict__ ls,
    float* __restrict__ ld, int H, int C)
{
    __shared__ float rs[256];
    __shared__ float rd[256];
    const int n = blockIdx.x;
    const int t = threadIdx.x;
    for (int hd = 0; hd < H; ++hd) {
        float ps = 0.0f, pd = 0.0f;
        for (int c = t; c < C; c += 256) {
            const float v = h[(size_t)n * (H * C) + hd * C + c];
            ps += v * a_s[hd * C + c];
            pd += v * a_d[hd * C + c];
        }
        rs[t] = ps;
        rd[t] = pd;
        __syncthreads();
        for (int s = 128; s > 0; s >>= 1) {
            if (t < s) { rs[t] += rs[t + s]; rd[t] += rd[t + s]; }
            __syncthreads();
        }
        if (t == 0) {
            ls[n * H + hd] = rs[0];
            ld[n * H + hd] = rd[0];
        }
        __syncthreads();
    }
}

// Monotone float<->int order mapping (for integer atomicMax segment-max).
__device__ __forceinline__ int f2ord(float f) {
    const int i = __float_as_int(f);
    return (i >= 0) ? i : (i ^ 0x7FFFFFFF);
}
__device__ __forceinline__ float ord2f(int i) {
    return __int_as_float((i >= 0) ? i : (i ^ 0x7FFFFFFF));
}

__global__ void fill_f32(float* __restrict__ p, float v, int n) {
    const int i = blockIdx.x * blockDim.x + threadIdx.x;
    if (i < n) p[i] = v;
}
__global__ void fill_i32(int* __restrict__ p, int v, int n) {
    const int i = blockIdx.x * blockDim.x + threadIdx.x;
    if (i < n) p[i] = v;
}

// e = leaky_relu(ls[src]+ld[dst]); segment max over dst via int atomicMax.
__global__ void edge_scores(
    const float* __restrict__ ls, const float* __restrict__ ld,
    const int* __restrict__ src, const int* __restrict__ dst,
    float* __restrict__ ev, int* __restrict__ m, int E, int H)
{
    const int i = blockIdx.x * blockDim.x + threadIdx.x;
    if (i >= E * H) return;
    const int e = i / H, hd = i - e * H;
    const int s = src[e], d = dst[e];
    float x = ls[s * H + hd] + ld[d * H + hd];
    x = (x > 0.0f) ? x : NEG_SLOPE * x;
    ev[i] = x;
    atomicMax(&m[d * H + hd], f2ord(x));
}

// ex = exp(e - m[dst]); segment sum z via atomicAdd. In-place on ev.
__global__ void edge_exp(
    float* __restrict__ ev, const int* __restrict__ dst,
    const int* __restrict__ m, float* __restrict__ z, int E, int H)
{
    const int i = blockIdx.x * blockDim.x + threadIdx.x;
    if (i >= E * H) return;
    const int e = i / H, hd = i - e * H;
    const int d = dst[e];
    const float ex = expf(ev[i] - ord2f(m[d * H + hd]));
    ev[i] = ex;
    atomicAdd(&z[d * H + hd], ex);
}

// num[dst] += ex[e,h] * h[src].  One block per edge; 192 threads x float4.
// A 4-wide chunk never straddles a head boundary (96 % 4 == 0).
__global__ __launch_bounds__(192) void aggregate(
    const float* __restrict__ h, const float* __restrict__ ex,
    const int* __restrict__ src, const int* __restrict__ dst,
    float* __restrict__ num, int H, int C)
{
    const int e = blockIdx.x;
    const int s = src[e], d = dst[e];
    const int col = threadIdx.x << 2;  // 0..764
    const int hd  = col / C;
    const float a = ex[e * H + hd];
    const float4 hv = *(const float4*)(h + (size_t)s * NFEAT + col);
    float* np = num + (size_t)d * NFEAT + col;
    atomicAdd(np + 0, hv.x * a);
    atomicAdd(np + 1, hv.y * a);
    atomicAdd(np + 2, hv.z * a);
    atomicAdd(np + 3, hv.w * a);
}

// out = num / (z + 1e-16) + bias, optional ELU. Safe in-place.
__global__ void finalize_kernel(
    const float* __restrict__ num, const float* __restrict__ z,
    const float* __restrict__ bias, float* __restrict__ out,
    int N, int H, int C, int apply_elu)
{
    const int i = blockIdx.x * blockDim.x + threadIdx.x;
    const int total = N * H * C;
    if (i >= total) return;
    const int F = H * C;
    const int n = i / F;
    const int f = i - n * F;
    const int hd = f / C;
    float v = num[i] / (z[n * H + hd] + 1e-16f) + bias[f];
    if (apply_elu) v = (v > 0.0f) ? v : (expf(v) - 1.0f);
    out[i] = v;
}

extern "C" void kernel_launch(void* const* d_in, const int* in_sizes, int n_in,
                              void* d_out, int out_size, void* d_ws, size_t ws_size,
                              hipStream_t stream)
{
    const float* x   = (const float*)d_in[0];
    const int*  eidx = (const int*)d_in[1];
    const float* W1  = (const float*)d_in[2];
    const float* as1 = (const float*)d_in[3];
    const float* ad1 = (const float*)d_in[4];
    const float* b1  = (const float*)d_in[5];
    const float* W2  = (const float*)d_in[6];
    const float* as2 = (const float*)d_in[7];
    const float* ad2 = (const float*)d_in[8];
    const float* b2  = (const float*)d_in[9];
    float* out = (float*)d_out;

    const int N = in_sizes[0] / NFEAT;   // 20000
    const int E = in_sizes[1] / 2;       // 340000
    const int* src = eidx;
    const int* dst = eidx + E;

    // Workspace layout:
    //   buf_h   : N*768  (h1, later h2)
    //   buf_num : N*768  (layer-1 accumulator, finalized in place -> layer-2 input)
    //   ls, ld, z : N*8 floats each ; m : N*8 ints ; ex : E*8 floats
    float* buf_h   = (float*)d_ws;
    float* buf_num = buf_h + (size_t)N * NFEAT;
    float* ls      = buf_num + (size_t)N * NFEAT;
    float* ld      = ls + (size_t)N * 8;
    float* z       = ld + (size_t)N * 8;
    int*   m       = (int*)(z + (size_t)N * 8);
    float* ex      = (float*)(m + (size_t)N * 8);

    const int ORD_NEG_INF = 0x807FFFFF;  // f2ord(-inf)
    const int BT = 256;
    const int mBlocks = N / ROWS;        // 250, exact

    // ------------------------- Layer 1 (H=8, C=96) -------------------------
    gemm_wmma_f32<<<mBlocks, 512, 0, stream>>>(x, W1, buf_h);
    logits_kernel<<<N, 256, 0, stream>>>(buf_h, as1, ad1, ls, ld, 8, 96);
    fill_f32<<<(N * 8 + BT - 1) / BT, BT, 0, stream>>>(z, 0.0f, N * 8);
    fill_i32<<<(N * 8 + BT - 1) / BT, BT, 0, stream>>>(m, ORD_NEG_INF, N * 8);
    fill_f32<<<(N * NFEAT + BT - 1) / BT, BT, 0, stream>>>(buf_num, 0.0f, N * NFEAT);
    edge_scores<<<(E * 8 + BT - 1) / BT, BT, 0, stream>>>(ls, ld, src, dst, ex, m, E, 8);
    edge_exp<<<(E * 8 + BT - 1) / BT, BT, 0, stream>>>(ex, dst, m, z, E, 8);
    aggregate<<<E, 192, 0, stream>>>(buf_h, ex, src, dst, buf_num, 8, 96);
    finalize_kernel<<<(N * NFEAT + BT - 1) / BT, BT, 0, stream>>>(
        buf_num, z, b1, buf_num, N, 8, 96, /*elu=*/1);

    // ------------------------- Layer 2 (H=1, C=768) ------------------------
    gemm_wmma_f32<<<mBlocks, 512, 0, stream>>>(buf_num, W2, buf_h);
    logits_kernel<<<N, 256, 0, stream>>>(buf_h, as2, ad2, ls, ld, 1, 768);
    fill_f32<<<(N + BT - 1) / BT, BT, 0, stream>>>(z, 0.0f, N);
    fill_i32<<<(N + BT - 1) / BT, BT, 0, stream>>>(m, ORD_NEG_INF, N);
    fill_f32<<<(N * NFEAT + BT - 1) / BT, BT, 0, stream>>>(out, 0.0f, N * NFEAT);
    edge_scores<<<(E + BT - 1) / BT, BT, 0, stream>>>(ls, ld, src, dst, ex, m, E, 1);
    edge_exp<<<(E + BT - 1) / BT, BT, 0, stream>>>(ex, dst, m, z, E, 1);
    aggregate<<<E, 192, 0, stream>>>(buf_h, ex, src, dst, out, 1, 768);
    finalize_kernel<<<(N * NFEAT + BT - 1) / BT, BT, 0, stream>>>(
        out, z, b2, out, N, 1, 768, /*elu=*/0);
}